// SpanPruner_85392539779528
// MI455X (gfx1250) — compile-verified
//
#include <hip/hip_runtime.h>
#include <hip/hip_bf16.h>
#include <stdint.h>

// ---------------------------------------------------------------------------
// SpanPruner on MI455X (gfx1250)
//   inputs:  embeddings [8,16384,256] f32, scores_a/b [8,16384,8] f32,
//            span_mask [8,16384] int, seq_length [8] int
//   outputs (concat, float-typed): top_indices [B,K], top_embeddings [B,K,256],
//            top_scores [B,K], top_mask [B,K];  K = out_size / (8*259)
//
// Kernel 1 (one 1024-thread block per batch row):
//   - compute masked max-tag-prob score per span (softmax over 8 tags, drop tag 0)
//   - pack (score,index) into sortable u64 keys, FULL-ROW bitonic sort in LDS
//     (16384 * 8B = 128KB -- fits only because CDNA5 has 320KB LDS/WGP)
//   - fill = max index among top-K; bitmap + popcount prefix-scan compaction
//     emits kept indices already ascending (== jnp.sort of top_indices)
// Kernel 2 (one 256-thread block per 16 output rows):
//   - Tensor Data Mover gather-mode descriptor pulls 16 embedding rows
//     (16-bit row indices) straight into LDS, s_wait_tensorcnt, copy out.
// ---------------------------------------------------------------------------

#define NSPANS 16384
#define DIM    256
#define NEG_FILL (-1e20f)

typedef __attribute__((ext_vector_type(4))) unsigned int u32x4;
typedef __attribute__((ext_vector_type(8))) int          i32x8;
typedef __attribute__((ext_vector_type(4))) int          i32x4;

__device__ __forceinline__ float tag_prob_max8(const float* s) {
  float4 a = ((const float4*)s)[0];
  float4 b = ((const float4*)s)[1];
  float m = fmaxf(fmaxf(fmaxf(a.x, a.y), fmaxf(a.z, a.w)),
                  fmaxf(fmaxf(b.x, b.y), fmaxf(b.z, b.w)));
  float e0 = __expf(a.x - m), e1 = __expf(a.y - m);
  float e2 = __expf(a.z - m), e3 = __expf(a.w - m);
  float e4 = __expf(b.x - m), e5 = __expf(b.y - m);
  float e6 = __expf(b.z - m), e7 = __expf(b.w - m);
  float denom = e0 + e1 + e2 + e3 + e4 + e5 + e6 + e7;
  float pm = fmaxf(fmaxf(fmaxf(e1, e2), fmaxf(e3, e4)), fmaxf(fmaxf(e5, e6), e7));
  return pm / denom;
}

__device__ __forceinline__ float masked_score(const float* __restrict__ sa,
                                              const float* __restrict__ sb,
                                              const int* __restrict__ mask,
                                              int b, int n) {
  size_t off = ((size_t)b * NSPANS + n) * 8;
  float p = fmaxf(tag_prob_max8(sa + off), tag_prob_max8(sb + off));
  return mask[(size_t)b * NSPANS + n] ? p : NEG_FILL;
}

__global__ void __launch_bounds__(1024)
span_prune_kernel(const float* __restrict__ sa, const float* __restrict__ sb,
                  const int* __restrict__ mask, const int* __restrict__ seq_len,
                  float* __restrict__ out, int K, int B) {
  extern __shared__ unsigned char smem[];
  unsigned long long* keys = (unsigned long long*)smem;              // 128 KB
  unsigned* bitmap = (unsigned*)(smem + 131072);                     // 2 KB
  unsigned* pc     = (unsigned*)(smem + 131072 + 2048);              // 2 KB
  int*      s_fill = (int*)(smem + 131072 + 4096);

  const int b = blockIdx.x;
  const int tid = threadIdx.x;
  const int nthr = blockDim.x;

  // ---- build sortable keys: (monotone(score) << 14) | (16383 - n) ----
  for (int n = tid; n < NSPANS; n += nthr) {
    float sc = masked_score(sa, sb, mask, b, n);
    unsigned u = __float_as_uint(sc);
    u = (u & 0x80000000u) ? ~u : (u | 0x80000000u);   // order-preserving map
    keys[n] = ((unsigned long long)u << 14) | (unsigned long long)(16383 - n);
  }
  __syncthreads();

  // ---- full-row bitonic sort, descending (top score first, lower n first) ----
  for (unsigned k2 = 2; k2 <= (unsigned)NSPANS; k2 <<= 1) {
    for (unsigned j = k2 >> 1; j > 0; j >>= 1) {
      for (unsigned i = tid; i < (unsigned)NSPANS; i += nthr) {
        unsigned ixj = i ^ j;
        if (ixj > i) {
          unsigned long long a = keys[i];
          unsigned long long c = keys[ixj];
          if ((a < c) == ((i & k2) == 0)) { keys[i] = c; keys[ixj] = a; }
        }
      }
      __syncthreads();
    }
  }

  // ---- num_keep, fill = max index among the K selected ----
  int seq = seq_len[b];
  int num_keep = seq * 2;
  if (num_keep < 1) num_keep = 1;
  if (num_keep > K) num_keep = K;

  if (tid == 0) *s_fill = 0;
  __syncthreads();
  int lm = 0;
  for (int i = tid; i < K; i += nthr)
    lm = max(lm, 16383 - (int)(keys[i] & 0x3FFFull));
  atomicMax(s_fill, lm);                      // ds_max on LDS

  // ---- bitmap of kept indices ----
  for (int t = tid; t < 512; t += nthr) bitmap[t] = 0u;
  __syncthreads();
  for (int i = tid; i < num_keep; i += nthr) {
    int idx = 16383 - (int)(keys[i] & 0x3FFFull);
    atomicOr(&bitmap[idx >> 5], 1u << (idx & 31));
  }
  __syncthreads();

  // ---- inclusive prefix scan of per-word popcounts (512 words) ----
  for (int t = tid; t < 512; t += nthr) pc[t] = __popc(bitmap[t]);
  __syncthreads();
  for (int off = 1; off < 512; off <<= 1) {
    unsigned v = 0;
    if (tid < 512 && tid >= off) v = pc[tid - off];
    __syncthreads();
    if (tid < 512) pc[tid] += v;
    __syncthreads();
  }
  int fill = *s_fill;

  float* out_idx = out;
  float* out_emb = out + (size_t)B * K;
  float* out_sc  = out_emb + (size_t)B * K * DIM;
  float* out_mk  = out_sc + (size_t)B * K;

  // ---- emit kept indices in ascending order via compaction ----
  for (int t = tid; t < 512; t += nthr) {
    unsigned w = bitmap[t];
    int p = (int)pc[t] - __popc(w);          // exclusive prefix
    while (w) {
      int bit = __ffs(w) - 1;
      int n = (t << 5) + bit;
      if (p < num_keep) {
        size_t o = (size_t)b * K + p;
        out_idx[o] = (float)n;
        out_sc[o]  = masked_score(sa, sb, mask, b, n);
        out_mk[o]  = mask[(size_t)b * NSPANS + n] ? 1.0f : 0.0f;
      }
      ++p;
      w &= w - 1;
    }
  }

  // ---- tail: positions [num_keep, K) are the fill index, mask 0 ----
  float fsc = masked_score(sa, sb, mask, b, fill);
  for (int p = num_keep + tid; p < K; p += nthr) {
    size_t o = (size_t)b * K + p;
    out_idx[o] = (float)fill;
    out_sc[o]  = fsc;
    out_mk[o]  = 0.0f;
  }
}

// ---------------------------------------------------------------------------
// TDM gather: 16 embedding rows (1 KB each) per descriptor -> LDS -> out.
// D# layout per CDNA5 ISA ch.8: group0 {count=1, gather_mode=1, 16-bit idx,
// lds_addr, global_addr, type=2}; group1 {data_size=4B, tensor_dim0=256,
// tensor_dim1=16384, tile_dim0=256, tile_dim1=nvalid, dim0_stride=256};
// groups 2-3 = sixteen packed 16-bit row indices.
// clang-23 toolchain builtin arity: (u32x4, i32x8, i32x4, i32x4, i32x8, cpol).
// ---------------------------------------------------------------------------
__global__ void __launch_bounds__(256)
embed_gather_kernel(const float* __restrict__ emb, const float* __restrict__ out_idx,
                    float* __restrict__ out_emb, int K, int B) {
  extern __shared__ unsigned char smem[];
  float* tile = (float*)smem;                  // 16 rows * 256 floats = 16 KB

  const int chunks = (K + 15) >> 4;
  const int b  = blockIdx.x / chunks;
  const int c  = blockIdx.x % chunks;
  const int k0 = c << 4;
  const int nvalid = min(16, K - k0);

  if (threadIdx.x < 32) {                      // wave 0 only (scalar branch)
    int idx[16];
#pragma unroll
    for (int r = 0; r < 16; ++r) {
      int rr = (r < nvalid) ? r : (nvalid - 1);
      idx[r] = (int)out_idx[(size_t)b * K + k0 + rr];
    }
    unsigned lds_off = (unsigned)(uintptr_t)tile;   // LDS byte offset
    unsigned long long ga =
        (unsigned long long)(uintptr_t)(emb + (size_t)b * NSPANS * DIM);

    u32x4 g0 = { 1u | (1u << 31),                       // count=1, gather_mode
                 lds_off,
                 (unsigned)(ga & 0xFFFFFFFFu),
                 (unsigned)((ga >> 32) & 0x01FFFFFFu) | (2u << 30) }; // type=2
    i32x8 g1 = { (int)(2u << 16),          // data_size = 4B
                 (int)(256u << 16),        // tensor_dim0 = 256 (low 16 @ [63:48])
                 (int)(16384u << 16),      // tensor_dim1 = 16384
                 (int)(256u << 16),        // tile_dim0 = 256
                 nvalid,                   // tile_dim1 = #valid gather indices
                 256,                      // tensor_dim0_stride = 256
                 0, 0 };
    i32x4 g2 = { idx[0] | (idx[1] << 16),  idx[2]  | (idx[3]  << 16),
                 idx[4] | (idx[5] << 16),  idx[6]  | (idx[7]  << 16) };
    i32x4 g3 = { idx[8] | (idx[9] << 16),  idx[10] | (idx[11] << 16),
                 idx[12] | (idx[13] << 16), idx[14] | (idx[15] << 16) };
    i32x8 g4 = { 0, 0, 0, 0, 0, 0, 0, 0 }; // 5th operand (uncharacterized), zero

    __builtin_amdgcn_tensor_load_to_lds(g0, g1, g2, g3, g4, 0);
    __builtin_amdgcn_s_wait_tensorcnt(0);
  }
  __syncthreads();

  const int r  = threadIdx.x >> 4;             // row within tile
  const int cg = threadIdx.x & 15;             // 16 threads per row
  if (r < nvalid) {
    const float4* src = (const float4*)(tile + r * DIM);
    float4* dst = (float4*)(out_emb + ((size_t)b * K + k0 + r) * DIM);
#pragma unroll
    for (int q = 0; q < 4; ++q) dst[cg + 16 * q] = src[cg + 16 * q];
  }
}

extern "C" void kernel_launch(void* const* d_in, const int* in_sizes, int n_in,
                              void* d_out, int out_size, void* d_ws, size_t ws_size,
                              hipStream_t stream) {
  (void)in_sizes; (void)n_in; (void)d_ws; (void)ws_size;
  const int B = 8;
  int K = out_size / (B * (DIM + 3));          // out = B*K*(1 + 256 + 1 + 1)
  if (K < 1) K = 1;

  const float* emb  = (const float*)d_in[0];
  const float* sa   = (const float*)d_in[1];
  const float* sb   = (const float*)d_in[2];
  const int*   mask = (const int*)d_in[3];
  const int*   seq  = (const int*)d_in[4];
  float* out = (float*)d_out;

  size_t lds1 = 131072 + 2048 + 2048 + 64;     // keys + bitmap + scan + fill
  span_prune_kernel<<<B, 1024, lds1, stream>>>(sa, sb, mask, seq, out, K, B);

  int chunks = (K + 15) / 16;
  size_t lds2 = 16 * DIM * sizeof(float);      // 16 KB tile
  embed_gather_kernel<<<B * chunks, 256, lds2, stream>>>(
      emb, out /* indices at base */, out + (size_t)B * K, K, B);
}